// _DecoderZX_46574625358364
// MI455X (gfx1250) — compile-verified
//
#include <hip/hip_runtime.h>
#include <cmath>

typedef __attribute__((ext_vector_type(2))) float v2f;
typedef __attribute__((ext_vector_type(8))) float v8f;

#define N_CELLS 1024
#define LAT     30
#define NGENE   10000
#define NBATCH  64
#define MAXTILES 128   // sum ceil(n_b/16) <= (1024 + 64*15)/16 = 124

// ---------------------------------------------------------------------------
// Kernel 1: counting-sort cells by batch, build M-tile table.
// ws layout (ints): counts[64] | offsets[65] | rank[1024] | order[1024]
//                   | tileBatch[128] | tileStart[128] | tileRows[128]
// ---------------------------------------------------------------------------
__global__ __launch_bounds__(1024) void sort_cells(const int* __restrict__ batch,
                                                   int* __restrict__ ws) {
    int* counts    = ws;
    int* offsets   = counts + NBATCH;
    int* rank      = offsets + NBATCH + 1;
    int* order     = rank + N_CELLS;
    int* tileBatch = order + N_CELLS;
    int* tileStart = tileBatch + MAXTILES;
    int* tileRows  = tileStart + MAXTILES;

    int t = threadIdx.x;
    if (t < NBATCH) counts[t] = 0;
    __syncthreads();
    int b = -1;
    if (t < N_CELLS) {
        b = batch[t];
        rank[t] = atomicAdd(&counts[b], 1);
    }
    __syncthreads();
    if (t == 0) {
        offsets[0] = 0;
        for (int i = 0; i < NBATCH; ++i) offsets[i + 1] = offsets[i] + counts[i];
        int nt = 0;
        for (int i = 0; i < NBATCH; ++i)
            for (int s = 0; s < counts[i]; s += 16) {
                tileBatch[nt] = i;
                tileStart[nt] = offsets[i] + s;
                tileRows[nt]  = min(16, counts[i] - s);
                ++nt;
            }
        for (; nt < MAXTILES; ++nt) { tileBatch[nt] = -1; tileStart[nt] = 0; tileRows[nt] = 0; }
    }
    __syncthreads();
    if (t < N_CELLS) order[offsets[b] + rank[t]] = t;
}

// ---------------------------------------------------------------------------
// Kernel 2: fused WMMA GEMM + bias (folded into C init) + softplus + scale.
// Block = 128 threads (4 wave32s); each wave owns one 16-gene tile; block
// covers 64 genes of one M-tile (16 cells of one batch).
// ---------------------------------------------------------------------------
__global__ __launch_bounds__(128) void decoder_wmma(
    const float* __restrict__ z,     // [N, L]
    const float* __restrict__ sf,    // [N, 1]
    const float* __restrict__ W,     // [L, G]
    const float* __restrict__ A,     // [NB, G, L]
    const float* __restrict__ bemb,  // [NB, G]
    float* __restrict__ out,         // [N, G]
    const int* __restrict__ ws) {
    const int* order     = ws + NBATCH + (NBATCH + 1) + N_CELLS;
    const int* tileBatch = order + N_CELLS;
    const int* tileStart = tileBatch + MAXTILES;
    const int* tileRows  = tileStart + MAXTILES;

    const int t   = blockIdx.y;
    const int bat = tileBatch[t];
    if (bat < 0) return;                 // uniform: whole block exits
    const int start = tileStart[t];
    const int rows  = tileRows[t];

    __shared__ int   cellIdx[16];
    __shared__ float sfv[16];
    __shared__ float zT[16][33];         // K padded to 32, stride 33 vs bank conflicts

    const int tid = threadIdx.x;
    if (tid < 16) {
        int cell = order[start + min(tid, rows - 1)];  // clamp keeps addresses valid
        cellIdx[tid] = cell;
        sfv[tid]     = sf[cell];
    }
    __syncthreads();
    for (int idx = tid; idx < 16 * 32; idx += 128) {
        int m = idx >> 5, k = idx & 31;
        float v = 0.f;
        if (m < rows && k < LAT) v = z[cellIdx[m] * LAT + k];
        zT[m][k] = v;                    // rows >= rows and K=30,31 zero-padded
    }
    __syncthreads();

    const int wid  = tid >> 5;
    const int lane = tid & 31;
    const int gbase = blockIdx.x * 64 + wid * 16;
    if (gbase >= NGENE) return;          // uniform per wave (G = 625*16 exactly)

    // A fragment: lane m<16 -> (M=m, K=k0,k0+1); lane m+16 -> (M=m, K=k0+2,k0+3)
    const int m_a   = lane & 15;
    const int khalf = (lane >> 4) << 1;  // 0 or 2
    // B fragment: lane -> column g; lane-half selects K row pair (same khalf)
    const int g = gbase + (lane & 15);
    const float* Arow = A + ((size_t)bat * NGENE + g) * LAT;  // A_emb[bat, g, :]

    // Bias depends only on N(=g): fold it into the accumulator (C/D layout
    // replicates N across all 8 VGPRs of a lane).
    const float bias = bemb[(size_t)bat * NGENE + g];
    v8f acc = {bias, bias, bias, bias, bias, bias, bias, bias};

#pragma unroll
    for (int kstep = 0; kstep < 8; ++kstep) {
        const int kb = kstep * 4 + khalf;
        // clamped indices keep every load in-bounds; tail lanes select 0 via
        // v_cndmask (no exec-mask branches between WMMAs)
        const int kc0 = (kb     < LAT) ? kb     : (LAT - 1);
        const int kc1 = (kb + 1 < LAT) ? kb + 1 : (LAT - 1);
        v2f a;
        a.x = zT[m_a][kc0] ;
        a.y = zT[m_a][kc1];
        float w0 = W[kc0 * NGENE + g] + Arow[kc0];
        float w1 = W[kc1 * NGENE + g] + Arow[kc1];
        v2f bb;
        bb.x = (kb     < LAT) ? w0 : 0.f;
        bb.y = (kb + 1 < LAT) ? w1 : 0.f;
        // D = A(16x4) * B(4x16) + C
        acc = __builtin_amdgcn_wmma_f32_16x16x4_f32(
            /*neg_a=*/false, a, /*neg_b=*/false, bb,
            /*c_mod=*/(short)0, acc, /*reuse_a=*/false, /*reuse_b=*/false);
    }

    const int mhalf = (lane >> 4) << 3;  // C/D: VGPR r holds M=r (lo) / M=r+8 (hi)
    if (rows == 16) {                    // fast path: full tile, no store masks
#pragma unroll
        for (int r = 0; r < 8; ++r) {
            const int m = mhalf + r;
            float x  = acc[r];
            // branchless stable softplus: max(x,0) + log1p(exp(-|x|))
            float sp = fmaxf(x, 0.f) + log1pf(expf(-fabsf(x)));
            out[(size_t)cellIdx[m] * NGENE + g] = sp * sfv[m];
        }
    } else {
#pragma unroll
        for (int r = 0; r < 8; ++r) {
            const int m = mhalf + r;
            float x  = acc[r];
            float sp = fmaxf(x, 0.f) + log1pf(expf(-fabsf(x)));
            float v  = sp * sfv[m];
            if (m < rows) out[(size_t)cellIdx[m] * NGENE + g] = v;
        }
    }
}

// ---------------------------------------------------------------------------
// Kernel 3: inverse_dispersion = exp(px_r)
// ---------------------------------------------------------------------------
__global__ void exp_px_r(const float* __restrict__ px_r, float* __restrict__ out) {
    int g = blockIdx.x * 256 + threadIdx.x;
    if (g < NGENE) out[g] = expf(px_r[g]);
}

extern "C" void kernel_launch(void* const* d_in, const int* in_sizes, int n_in,
                              void* d_out, int out_size, void* d_ws, size_t ws_size,
                              hipStream_t stream) {
    const float* z     = (const float*)d_in[0];
    const int*   batch = (const int*)d_in[1];
    const float* sf    = (const float*)d_in[2];
    const float* W     = (const float*)d_in[3];
    const float* A     = (const float*)d_in[4];
    const float* bemb  = (const float*)d_in[5];
    const float* px_r  = (const float*)d_in[6];
    float* out = (float*)d_out;
    int*   ws  = (int*)d_ws;   // needs ~10 KB

    sort_cells<<<1, 1024, 0, stream>>>(batch, ws);

    dim3 grid((NGENE + 63) / 64, MAXTILES);
    decoder_wmma<<<grid, 128, 0, stream>>>(z, sf, W, A, bemb, out, ws);

    exp_px_r<<<(NGENE + 255) / 256, 256, 0, stream>>>(px_r, out + (size_t)N_CELLS * NGENE);
}